// SAGE_4105988735602
// MI455X (gfx1250) — compile-verified
//
#include <hip/hip_runtime.h>

#define NN 100000
#define EE 1600000
#define KDIM 128   // input dim of every layer
#define HID 128
#define OUTD 64

typedef __attribute__((ext_vector_type(2))) float v2f;
typedef __attribute__((ext_vector_type(8))) float v8f;

// ---------------------------------------------------------------- utilities
__global__ void zero_f32(float* __restrict__ p, long long n) {
    long long i = (long long)blockIdx.x * blockDim.x + threadIdx.x;
    long long stride = (long long)gridDim.x * blockDim.x;
    for (; i < n; i += stride) p[i] = 0.0f;
}

// degree count: cnt[dst] += 1
__global__ void count_edges(const long long* __restrict__ ei, float* __restrict__ cnt) {
    int e = blockIdx.x * 256 + threadIdx.x;
    if (e < EE) {
        int d = (int)ei[EE + e];
        unsafeAtomicAdd(&cnt[d], 1.0f);
    }
}

__global__ void invert_cnt(float* __restrict__ cnt) {
    int i = blockIdx.x * 256 + threadIdx.x;
    if (i < NN) cnt[i] = 1.0f / fmaxf(cnt[i], 1.0f);
}

// fold bias + BN into per-column affine: out = gemm*scale + shift
__global__ void prep_affine(const float* __restrict__ bl, const float* __restrict__ g,
                            const float* __restrict__ b,  const float* __restrict__ m,
                            const float* __restrict__ v,  float* __restrict__ scale,
                            float* __restrict__ shift, int nc, int hasBN) {
    int c = blockIdx.x * blockDim.x + threadIdx.x;
    if (c < nc) {
        if (hasBN) {
            float s = g[c] * rsqrtf(v[c] + 1e-5f);
            scale[c] = s;
            shift[c] = (bl[c] - m[c]) * s + b[c];
        } else {
            scale[c] = 1.0f;
            shift[c] = bl[c];
        }
    }
}

// ------------------------------------------------- edge scatter aggregation
// one wave (32 lanes) per edge; each lane moves 4 floats (128 total)
__global__ void aggregate(const float* __restrict__ x, const long long* __restrict__ ei,
                          float* __restrict__ agg) {
    long long t = (long long)blockIdx.x * 256 + threadIdx.x;   // < EE*32 exactly
    int e  = (int)(t >> 5);
    int ln = (int)(t & 31);
    long long s = ei[e];
    long long d = ei[EE + e];
    const float4 val = *(const float4*)(x + s * (long long)KDIM + ln * 4);
    float* dst = agg + d * (long long)KDIM + ln * 4;
    unsafeAtomicAdd(dst + 0, val.x);
    unsafeAtomicAdd(dst + 1, val.y);
    unsafeAtomicAdd(dst + 2, val.z);
    unsafeAtomicAdd(dst + 3, val.w);
}

// ------------------------------------------------------- fused WMMA GEMM
// out[n, :] = act( ((agg[n]*invc[n]) @ Wl + xin[n] @ Wr) * scale + shift )
// K = 128. One wave -> 16-row tile, full width NC. fp32 WMMA 16x16x4.
template<int NC, bool RELU>
__global__ __launch_bounds__(256) void gemm_fused(
        const float* __restrict__ xin, const float* __restrict__ agg,
        const float* __restrict__ invc,
        const float* __restrict__ Wl,  const float* __restrict__ Wr,
        const float* __restrict__ scale, const float* __restrict__ shift,
        float* __restrict__ out) {
    // weights swizzled: pair p = k>>1 ; Bs[mat][p][n][k&1]
    __shared__ float Bs[2 * (KDIM / 2) * 2 * NC];
    __shared__ float s_scale[NC];
    __shared__ float s_shift[NC];

    const int tid = threadIdx.x;
    for (int i = tid; i < KDIM * NC; i += 256) {
        int k = i / NC;
        int n = i - k * NC;
        int idx = (k >> 1) * (2 * NC) + n * 2 + (k & 1);
        Bs[idx]                     = Wl[i];
        Bs[(KDIM / 2) * 2 * NC + idx] = Wr[i];
    }
    for (int i = tid; i < NC; i += 256) { s_scale[i] = scale[i]; s_shift[i] = shift[i]; }
    __syncthreads();

    const int wave  = tid >> 5;
    const int lane  = tid & 31;
    const int lhalf = lane & 15;
    const int hi    = lane >> 4;           // 0: K rows {k,k+1}, 1: {k+2,k+3}
    const int numTiles = NN / 16;          // 6250

    for (int tile = blockIdx.x * 8 + wave; tile < numTiles; tile += gridDim.x * 8) {
        const int arow = tile * 16 + lhalf;       // this lane's A row
        const float rinv = invc[arow];

        v8f c[NC / 16];
#pragma unroll
        for (int t = 0; t < NC / 16; ++t) c[t] = v8f{0, 0, 0, 0, 0, 0, 0, 0};

        for (int phase = 0; phase < 2; ++phase) {
            const float* A   = phase ? xin : agg;
            const float  scl = phase ? 1.0f : rinv;
            const int    boff = phase * (KDIM / 2) * 2 * NC;
            const float* Arow = A + (long long)arow * KDIM + hi * 2;

            for (int k0 = 0; k0 < KDIM; k0 += 4) {
                v2f a = *(const v2f*)(Arow + k0);
                v2f av; av.x = a.x * scl; av.y = a.y * scl;
                const int prow = boff + ((k0 >> 1) + hi) * (2 * NC);
#pragma unroll
                for (int t = 0; t < NC / 16; ++t) {
                    v2f bv = *(const v2f*)&Bs[prow + (t * 16 + lhalf) * 2];
                    c[t] = __builtin_amdgcn_wmma_f32_16x16x4_f32(
                        false, av, false, bv, (short)0, c[t], false, false);
                }
            }
        }

        // epilogue: C layout -> VGPR r holds M = r + hi*8, N = lhalf
#pragma unroll
        for (int t = 0; t < NC / 16; ++t) {
            const int ocol = t * 16 + lhalf;
            const float sc = s_scale[ocol];
            const float sh = s_shift[ocol];
#pragma unroll
            for (int r = 0; r < 8; ++r) {
                int orow = tile * 16 + r + hi * 8;
                float vv = c[t][r] * sc + sh;
                if (RELU) vv = fmaxf(vv, 0.0f);
                out[(long long)orow * NC + ocol] = vv;
            }
        }
    }
}

// ---------------------------------------------------------------- launcher
extern "C" void kernel_launch(void* const* d_in, const int* in_sizes, int n_in,
                              void* d_out, int out_size, void* d_ws, size_t ws_size,
                              hipStream_t stream) {
    const float*     x   = (const float*)d_in[0];
    const long long* ei  = (const long long*)d_in[1];
    const float* Wl0 = (const float*)d_in[2];
    const float* Wr0 = (const float*)d_in[3];
    const float* bl0 = (const float*)d_in[4];
    const float* Wl1 = (const float*)d_in[5];
    const float* Wr1 = (const float*)d_in[6];
    const float* bl1 = (const float*)d_in[7];
    const float* Wl2 = (const float*)d_in[8];
    const float* Wr2 = (const float*)d_in[9];
    const float* bl2 = (const float*)d_in[10];
    const float* g0 = (const float*)d_in[11];
    const float* b0 = (const float*)d_in[12];
    const float* m0 = (const float*)d_in[13];
    const float* v0 = (const float*)d_in[14];
    const float* g1 = (const float*)d_in[15];
    const float* b1 = (const float*)d_in[16];
    const float* m1 = (const float*)d_in[17];
    const float* v1 = (const float*)d_in[18];
    float* out = (float*)d_out;

    float* f   = (float*)d_ws;
    float* inv = f;                                   // N
    float* agg = inv + NN;                            // N*128
    float* h1  = agg + (long long)NN * KDIM;          // N*128
    float* h2  = h1  + (long long)NN * KDIM;          // N*128
    float* aff = h2  + (long long)NN * KDIM;          // 6*128
    float* sc0 = aff +   0; float* sh0 = aff + 128;
    float* sc1 = aff + 256; float* sh1 = aff + 384;
    float* sc2 = aff + 512; float* sh2 = aff + 640;

    const long long aggN = (long long)NN * KDIM;
    const int gemmGrid = (NN / 16 + 7) / 8;           // 782 blocks, 8 waves each
    const int aggGrid  = (int)((long long)EE * 32 / 256);  // exact 200000

    // degrees (shared by all 3 layers)
    zero_f32<<<1024, 256, 0, stream>>>(inv, NN);
    count_edges<<<(EE + 255) / 256, 256, 0, stream>>>(ei, inv);
    invert_cnt<<<(NN + 255) / 256, 256, 0, stream>>>(inv);

    // fold bias/BN into per-column affine
    prep_affine<<<1, 128, 0, stream>>>(bl0, g0, b0, m0, v0, sc0, sh0, HID, 1);
    prep_affine<<<1, 128, 0, stream>>>(bl1, g1, b1, m1, v1, sc1, sh1, HID, 1);
    prep_affine<<<1, 128, 0, stream>>>(bl2, nullptr, nullptr, nullptr, nullptr, sc2, sh2, OUTD, 0);

    // ---- layer 0: x -> h1
    zero_f32<<<4096, 256, 0, stream>>>(agg, aggN);
    aggregate<<<aggGrid, 256, 0, stream>>>(x, ei, agg);
    gemm_fused<HID, true><<<gemmGrid, 256, 0, stream>>>(x, agg, inv, Wl0, Wr0, sc0, sh0, h1);

    // ---- layer 1: h1 -> h2
    zero_f32<<<4096, 256, 0, stream>>>(agg, aggN);
    aggregate<<<aggGrid, 256, 0, stream>>>(h1, ei, agg);
    gemm_fused<HID, true><<<gemmGrid, 256, 0, stream>>>(h1, agg, inv, Wl1, Wr1, sc1, sh1, h2);

    // ---- layer 2: h2 -> out
    zero_f32<<<4096, 256, 0, stream>>>(agg, aggN);
    aggregate<<<aggGrid, 256, 0, stream>>>(h2, ei, agg);
    gemm_fused<OUTD, false><<<gemmGrid, 256, 0, stream>>>(h2, agg, inv, Wl2, Wr2, sc2, sh2, out);
}